// NCEAverage_13374528159993
// MI455X (gfx1250) — compile-verified
//
#include <hip/hip_runtime.h>
#include <hip/hip_bf16.h>

typedef float v2f __attribute__((ext_vector_type(2)));
typedef float v8f __attribute__((ext_vector_type(8)));

#define BATCH 128
#define DIM 128
#define XS_STRIDE 132            // padded row stride (floats) -> conflict-free ds_load_b64
#define WAVES_PER_BLOCK 8
#define LOG2E_OVER_T 14.4269504088896341f   // log2(e)/T, T = 0.1 : exp(s/T) = exp2(s*this)
#define MAX_BLOCKS 1024

static __device__ __forceinline__ float fast_exp_scaled(float s) {
    // exp(s / T) == exp2(s * log2(e) / T); v_exp_f32 is natively exp2.
    return __builtin_amdgcn_exp2f(s * LOG2E_OVER_T);
}

// Kernel 1: scores = x @ memory^T via V_WMMA_F32_16X16X4_F32.
// Each wave owns a 32-column span (two 16-wide N-tiles) so every A-fragment
// LDS load feeds two WMMAs (A is N-invariant). exp + partial-Z per block.
__global__ __launch_bounds__(256) void nce_z_kernel(
    const float* __restrict__ x, const float* __restrict__ mem,
    float* __restrict__ partial, int nspans)
{
    __shared__ float xs[BATCH * XS_STRIDE];
    __shared__ float zpart[WAVES_PER_BLOCK][BATCH];

    const int tid = threadIdx.x;

    // Stage x[128][128] into LDS with padded stride (float4 stores, 16B aligned: 132*4=528%16==0)
    for (int i = tid; i < (BATCH * DIM) / 4; i += 256) {
        const int row = i >> 5;       // 32 float4 per row
        const int c4  = i & 31;
        const float4 v = reinterpret_cast<const float4*>(x)[i];
        *reinterpret_cast<float4*>(&xs[row * XS_STRIDE + c4 * 4]) = v;
    }
    __syncthreads();

    const int lane = tid & 31;
    const int wave = tid >> 5;
    const int l    = lane & 15;       // N index within tile (B/C/D), M row group (A)
    const int h    = lane >> 4;       // half select: K pair (A/B), M+8 (C/D)

    const int gw      = blockIdx.x * WAVES_PER_BLOCK + wave;
    const int gwTotal = gridDim.x * WAVES_PER_BLOCK;

    // Running exp-sums, same VGPR layout as C/D: S[mt][r] ~ m = mt*16 + r + 8*h
    v8f S[8];
    #pragma unroll
    for (int mt = 0; mt < 8; ++mt)
        #pragma unroll
        for (int i = 0; i < 8; ++i) S[mt][i] = 0.0f;

    for (int span = gw; span < nspans; span += gwTotal) {
        const long long n0 = (long long)span * 32;
        // B fragment source: B[k][n] = mem[n][k]; per lane n = n0+l (+16), cols k+2h, k+2h+1
        const float* brow0 = mem + (n0 + l) * (long long)DIM + 2 * h;
        const float* brow1 = brow0 + 16LL * DIM;

        v8f acc0[8], acc1[8];
        #pragma unroll
        for (int mt = 0; mt < 8; ++mt)
            #pragma unroll
            for (int i = 0; i < 8; ++i) { acc0[mt][i] = 0.0f; acc1[mt][i] = 0.0f; }

        #pragma unroll 2
        for (int kk = 0; kk < DIM / 4; ++kk) {          // k = 4*kk
            const v2f b0 = *reinterpret_cast<const v2f*>(brow0 + kk * 4);
            const v2f b1 = *reinterpret_cast<const v2f*>(brow1 + kk * 4);
            #pragma unroll
            for (int mt = 0; mt < 8; ++mt) {
                // A 16x4 f32 layout: lanes 0-15 M=row, VGPR0/1 = K = k+2h, k+2h+1
                const v2f a = *reinterpret_cast<const v2f*>(
                    &xs[(mt * 16 + l) * XS_STRIDE + kk * 4 + 2 * h]);
                acc0[mt] = __builtin_amdgcn_wmma_f32_16x16x4_f32(
                    false, a, false, b0, (short)0, acc0[mt], false, false);
                acc1[mt] = __builtin_amdgcn_wmma_f32_16x16x4_f32(
                    false, a, false, b1, (short)0, acc1[mt], false, false);
            }
        }

        #pragma unroll
        for (int mt = 0; mt < 8; ++mt)
            #pragma unroll
            for (int r = 0; r < 8; ++r)
                S[mt][r] += fast_exp_scaled(acc0[mt][r]) + fast_exp_scaled(acc1[mt][r]);
    }

    // Reduce over the 16 N-columns (lanes within each half), per (mt, r)
    #pragma unroll
    for (int mt = 0; mt < 8; ++mt) {
        #pragma unroll
        for (int r = 0; r < 8; ++r) {
            float t = S[mt][r];
            t += __shfl_xor(t, 1);
            t += __shfl_xor(t, 2);
            t += __shfl_xor(t, 4);
            t += __shfl_xor(t, 8);
            if ((lane & 15) == 0)
                zpart[wave][mt * 16 + r + 8 * h] = t;  // lane 0 -> M=r, lane 16 -> M=r+8
        }
    }
    __syncthreads();

    // Deterministic per-block reduction over the 8 waves
    if (tid < BATCH) {
        float z = 0.0f;
        #pragma unroll
        for (int w = 0; w < WAVES_PER_BLOCK; ++w) z += zpart[w][tid];
        partial[(long long)blockIdx.x * BATCH + tid] = z;
    }
}

// Kernel 2: deterministic block-partial reduction + gathered numerator.
__global__ __launch_bounds__(128) void nce_finalize(
    const float* __restrict__ partial, int nblocks,
    const float* __restrict__ x, const float* __restrict__ mem,
    const int* __restrict__ y, float* __restrict__ out,
    int nrows, int nspans)
{
    const int b = threadIdx.x;
    if (b >= BATCH) return;

    float z = 0.0f;
    for (int i = 0; i < nblocks; ++i) z += partial[(long long)i * BATCH + b];

    // Tail rows not covered by full 32-wide spans (none for 400000, kept for generality)
    for (int n = nspans * 32; n < nrows; ++n) {
        const float* m = mem + (long long)n * DIM;
        float d = 0.0f;
        for (int k = 0; k < DIM; ++k) d += x[b * DIM + k] * m[k];
        z += fast_exp_scaled(d);
    }

    const float* w = mem + (long long)y[b] * DIM;
    float d = 0.0f;
    for (int k = 0; k < DIM; ++k) d += x[b * DIM + k] * w[k];
    out[b] = fast_exp_scaled(d) / z;
}

extern "C" void kernel_launch(void* const* d_in, const int* in_sizes, int n_in,
                              void* d_out, int out_size, void* d_ws, size_t ws_size,
                              hipStream_t stream) {
    const float* x   = (const float*)d_in[0];
    const int*   y   = (const int*)d_in[1];
    const float* mem = (const float*)d_in[2];
    float*       out = (float*)d_out;

    const int nrows  = in_sizes[2] / DIM;     // 400000
    const int nspans = nrows / 32;            // 12500 (exact)

    // Size grid from available workspace (deterministic: ws_size is fixed per run)
    long long maxb = (long long)(ws_size / (BATCH * sizeof(float)));
    int nblocks = (int)(maxb < 1 ? 1 : (maxb > MAX_BLOCKS ? MAX_BLOCKS : maxb));

    float* partial = (float*)d_ws;

    nce_z_kernel<<<nblocks, 256, 0, stream>>>(x, mem, partial, nspans);
    nce_finalize<<<1, 128, 0, stream>>>(partial, nblocks, x, mem, y, out, nrows, nspans);
}